// GraphTransformerEncode_89859305767438
// MI455X (gfx1250) — compile-verified
//
#include <hip/hip_runtime.h>
#include <hip/hip_bf16.h>

typedef __attribute__((ext_vector_type(2))) float v2f;
typedef __attribute__((ext_vector_type(8))) float v8f;

#define D_MODEL 128
#define NHEAD   8
#define DFF     512

// ---------------------------------------------------------------------------
// zero scratch (agg + ks_sum)
// ---------------------------------------------------------------------------
__global__ void zero_kernel(float* __restrict__ p, int n) {
    int i = blockIdx.x * blockDim.x + threadIdx.x;
    if (i < n) p[i] = 0.0f;
}

// ---------------------------------------------------------------------------
// LayerNorm over D=128: one wave32 per row, float4 per lane.
// ---------------------------------------------------------------------------
__global__ __launch_bounds__(256) void ln_kernel(const float* __restrict__ in,
                                                 const float* __restrict__ g,
                                                 const float* __restrict__ b,
                                                 float* __restrict__ out,
                                                 int rows) {
    int w    = (blockIdx.x * blockDim.x + threadIdx.x) >> 5;
    int lane = threadIdx.x & 31;
    if (w >= rows) return;
    const float4 f = *(const float4*)(in + w * D_MODEL + lane * 4);
    float s = f.x + f.y + f.z + f.w;
    #pragma unroll
    for (int m = 16; m; m >>= 1) s += __shfl_xor(s, m, 32);
    float mu = s * (1.0f / 128.0f);
    float dx = f.x - mu, dy = f.y - mu, dz = f.z - mu, dw = f.w - mu;
    float vv = dx * dx + dy * dy + dz * dz + dw * dw;
    #pragma unroll
    for (int m = 16; m; m >>= 1) vv += __shfl_xor(vv, m, 32);
    float inv = rsqrtf(vv * (1.0f / 128.0f) + 1e-6f);
    const float4 gg = *(const float4*)(g + lane * 4);
    const float4 bb = *(const float4*)(b + lane * 4);
    float4 o;
    o.x = dx * inv * gg.x + bb.x;
    o.y = dy * inv * gg.y + bb.y;
    o.z = dz * inv * gg.z + bb.z;
    o.w = dw * inv * gg.w + bb.w;
    *(float4*)(out + w * D_MODEL + lane * 4) = o;
}

// ---------------------------------------------------------------------------
// fp32 WMMA GEMM, compile-time shapes: C[M,NC] = A[M,K] @ B[K,NC] + bias
// (+ optional residual). One wave32 per 16x16 tile, v_wmma_f32_16x16x4_f32.
// 32-bit indexing so the backend can use saddr + voffset addressing.
// ---------------------------------------------------------------------------
template <int K, int NC, bool RESID>
__global__ __launch_bounds__(256) void gemm_kernel(const float* __restrict__ A,
                                                   const float* __restrict__ B,
                                                   const float* __restrict__ bias,
                                                   const float* __restrict__ resid,
                                                   float* __restrict__ C,
                                                   int M) {
    int tid  = blockIdx.x * blockDim.x + threadIdx.x;
    int tile = tid >> 5;
    int lane = threadIdx.x & 31;
    const int ntn = NC >> 4;
    int tot  = (M >> 4) * ntn;
    if (tile >= tot) return;                 // wave-uniform: EXEC stays all-ones
    int tm = tile / ntn;
    int tn = tile - tm * ntn;
    int lo = lane & 15, hi = lane >> 4;
    int m  = tm * 16 + lo;
    int n  = tn * 16 + lo;
    const float* Ap = A + m * K + 2 * hi;    // lane's k-pair base within row
    const float* Bp = B + 2 * hi * NC + n;   // lane's k-pair base within col
    __builtin_prefetch(Ap, 0, 0);            // global_prefetch_b8 (cacheline)
    v8f acc = {};
    #pragma unroll 8
    for (int kk = 0; kk < K; kk += 4) {
        v2f a, bfr;
        a.x   = Ap[kk];
        a.y   = Ap[kk + 1];
        bfr.x = Bp[kk * NC];
        bfr.y = Bp[kk * NC + NC];
        acc = __builtin_amdgcn_wmma_f32_16x16x4_f32(false, a, false, bfr,
                                                    (short)0, acc, false, false);
    }
    float bv = bias[n];
    #pragma unroll
    for (int j = 0; j < 8; ++j) {
        int row = tm * 16 + j + 8 * hi;
        int gi  = row * NC + n;
        float val = acc[j] + bv;
        if (RESID) val += resid[gi];
        C[gi] = val;
    }
}

// ---------------------------------------------------------------------------
// Column sums of k [N,128] -> ks_sum[128]
// ---------------------------------------------------------------------------
__global__ __launch_bounds__(128) void colsum_kernel(const float* __restrict__ k,
                                                     float* __restrict__ ks_sum,
                                                     int rows) {
    int col = threadIdx.x;
    float s = 0.0f;
    for (int r = blockIdx.x; r < rows; r += gridDim.x)
        s += k[r * D_MODEL + col];
    atomicAdd(&ks_sum[col], s);
}

// ---------------------------------------------------------------------------
// denom[n,h] = dot16(q[n,h,:], ks_sum[h,:])
// ---------------------------------------------------------------------------
__global__ __launch_bounds__(256) void denom_kernel(const float* __restrict__ q,
                                                    const float* __restrict__ ks_sum,
                                                    float* __restrict__ denom,
                                                    int rows) {
    int idx = blockIdx.x * blockDim.x + threadIdx.x;
    if (idx >= rows * NHEAD) return;
    int n = idx >> 3, h = idx & 7;
    const float* qp = q + n * D_MODEL + h * 16;
    const float* kp = ks_sum + h * 16;
    float s = 0.0f;
    #pragma unroll
    for (int d = 0; d < 16; ++d) s += qp[d] * kp[d];
    denom[idx] = s;
}

// ---------------------------------------------------------------------------
// Edge kernel: one wave32 per edge. Per-head (16-wide) dot via shfl_xor,
// attn write + fp32 atomic scatter into agg. All gathers hit the 192MB L2
// (q/k/v arrays are 25.6MB each).
// ---------------------------------------------------------------------------
__global__ __launch_bounds__(256) void edge_kernel(const int* __restrict__ eidx,
                                                   const int* __restrict__ spv,
                                                   const int* __restrict__ erel,
                                                   const float* __restrict__ rel_emb,
                                                   const float* __restrict__ sp_emb,
                                                   const float* __restrict__ q,
                                                   const float* __restrict__ k,
                                                   const float* __restrict__ v,
                                                   const float* __restrict__ denom,
                                                   float* __restrict__ agg,
                                                   float* __restrict__ attn,
                                                   int E) {
    int w    = (blockIdx.x * blockDim.x + threadIdx.x) >> 5;
    int lane = threadIdx.x & 31;
    if (w >= E) return;
    int   r  = eidx[w];
    int   c  = eidx[E + w];
    int   ri = erel[w];
    float sp = sp_emb[spv[w]];
    const float4 q4 = *(const float4*)(q + c * D_MODEL + lane * 4);
    const float4 k4 = *(const float4*)(k + r * D_MODEL + lane * 4);
    const float4 r4 = *(const float4*)(rel_emb + ri * D_MODEL + lane * 4);
    const float4 v4 = *(const float4*)(v + r * D_MODEL + lane * 4);
    float qex = q4.x + r4.x, qey = q4.y + r4.y, qez = q4.z + r4.z, qew = q4.w + r4.w;
    float kex = k4.x + r4.x, key = k4.y + r4.y, kez = k4.z + r4.z, kew = k4.w + r4.w;
    float num = qex * kex + qey * key + qez * kez + qew * kew;
    // reduce over the 4 lanes of each head (16 floats / head)
    num += __shfl_xor(num, 1, 32);
    num += __shfl_xor(num, 2, 32);
    int h = lane >> 2;
    // 1/sqrt(sqrt(DEPTH=16)) = 0.5
    num = num * 0.5f + sp;
    float a = num / denom[c * NHEAD + h];
    if ((lane & 3) == 0) attn[w * NHEAD + h] = a;
    float* ap = agg + c * D_MODEL + lane * 4;
    atomicAdd(ap + 0, a * v4.x);
    atomicAdd(ap + 1, a * v4.y);
    atomicAdd(ap + 2, a * v4.z);
    atomicAdd(ap + 3, a * v4.w);
}

// ---------------------------------------------------------------------------
// Fused FFN: per 16-row tile, t = relu(hln@W1+b1) in LDS, out2 = t@W2+b2+out1.
// All matmuls via v_wmma_f32_16x16x4_f32. 8 waves / block, 33KB LDS.
// ---------------------------------------------------------------------------
#define TPAD 520
__global__ __launch_bounds__(256) void ffn_kernel(const float* __restrict__ hln,
                                                  const float* __restrict__ W1,
                                                  const float* __restrict__ b1,
                                                  const float* __restrict__ W2,
                                                  const float* __restrict__ b2,
                                                  const float* __restrict__ out1,
                                                  float* __restrict__ out2) {
    __shared__ float t[16 * TPAD];
    const int lane = threadIdx.x & 31;
    const int wave = threadIdx.x >> 5;
    const int m0   = blockIdx.x * 16;
    const int lo   = lane & 15;
    const int hi   = lane >> 4;
    // Phase A: t[16,512] = relu(hln_tile[16,128] @ W1 + b1)
    {
        const float* Ap = hln + (m0 + lo) * D_MODEL + 2 * hi;
        __builtin_prefetch(Ap, 0, 0);
        #pragma unroll
        for (int it = 0; it < 4; ++it) {
            int nt = wave + it * 8;
            int n  = nt * 16 + lo;
            const float* Bp = W1 + 2 * hi * DFF + n;
            v8f acc = {};
            #pragma unroll 8
            for (int kk = 0; kk < D_MODEL; kk += 4) {
                v2f a, bfr;
                a.x   = Ap[kk];
                a.y   = Ap[kk + 1];
                bfr.x = Bp[kk * DFF];
                bfr.y = Bp[kk * DFF + DFF];
                acc = __builtin_amdgcn_wmma_f32_16x16x4_f32(false, a, false, bfr,
                                                            (short)0, acc, false, false);
            }
            float bias = b1[n];
            #pragma unroll
            for (int j = 0; j < 8; ++j) {
                int row = j + 8 * hi;
                float val = acc[j] + bias;
                t[row * TPAD + n] = val > 0.0f ? val : 0.0f;
            }
        }
    }
    __syncthreads();
    // Phase B: out2_tile[16,128] = t @ W2 + b2 + out1_tile
    {
        const int n = wave * 16 + lo;
        const float* Bp = W2 + 2 * hi * D_MODEL + n;
        const float* tp = t + lo * TPAD + 2 * hi;
        v8f acc = {};
        #pragma unroll 8
        for (int kk = 0; kk < DFF; kk += 4) {
            v2f a, bfr;
            a.x   = tp[kk];
            a.y   = tp[kk + 1];
            bfr.x = Bp[kk * D_MODEL];
            bfr.y = Bp[kk * D_MODEL + D_MODEL];
            acc = __builtin_amdgcn_wmma_f32_16x16x4_f32(false, a, false, bfr,
                                                        (short)0, acc, false, false);
        }
        float bias = b2[n];
        #pragma unroll
        for (int j = 0; j < 8; ++j) {
            int row = j + 8 * hi;
            int gi  = (m0 + row) * D_MODEL + n;
            out2[gi] = acc[j] + bias + out1[gi];
        }
    }
}

// ---------------------------------------------------------------------------
// Launch
// ---------------------------------------------------------------------------
extern "C" void kernel_launch(void* const* d_in, const int* in_sizes, int n_in,
                              void* d_out, int out_size, void* d_ws, size_t ws_size,
                              hipStream_t stream) {
    const float* feature = (const float*)d_in[0];
    const int*   eidx    = (const int*)d_in[1];
    const int*   spv     = (const int*)d_in[2];
    const int*   erel    = (const int*)d_in[3];
    const float* rel_emb = (const float*)d_in[4];
    const float* sp_emb  = (const float*)d_in[5];
    const float* Wq = (const float*)d_in[6];  const float* bq = (const float*)d_in[7];
    const float* Wk = (const float*)d_in[8];  const float* bk = (const float*)d_in[9];
    const float* Wv = (const float*)d_in[10]; const float* bv = (const float*)d_in[11];
    const float* Wd = (const float*)d_in[12]; const float* bd = (const float*)d_in[13];
    const float* W1 = (const float*)d_in[14]; const float* b1 = (const float*)d_in[15];
    const float* W2 = (const float*)d_in[16]; const float* b2 = (const float*)d_in[17];
    const float* g1 = (const float*)d_in[18]; const float* be1 = (const float*)d_in[19];
    const float* g2 = (const float*)d_in[20]; const float* be2 = (const float*)d_in[21];

    const int N  = in_sizes[0] / D_MODEL;   // 50000
    const int E  = in_sizes[1] / 2;         // 800000
    const int ND = N * D_MODEL;             // 6.4M, fits int

    float* ws     = (float*)d_ws;
    float* x      = ws;                     // LN1 output; reused as hln later
    float* q      = x    + ND;
    float* k      = q    + ND;
    float* v      = k    + ND;
    float* out1   = v    + ND;
    float* denom  = out1 + ND;              // N*8
    float* agg    = denom + N * NHEAD;
    float* ks_sum = agg + ND;               // 128 floats, contiguous after agg

    float* out2 = (float*)d_out;
    float* attn = out2 + ND;

    // 1. zero agg + ks_sum
    {
        int n = ND + D_MODEL;
        zero_kernel<<<(n + 255) / 256, 256, 0, stream>>>(agg, n);
    }
    // 2. x = LN(feature)
    ln_kernel<<<(N + 7) / 8, 256, 0, stream>>>(feature, g1, be1, x, N);
    // 3. q/k/v projections (WMMA)
    {
        int tiles  = (N / 16) * (D_MODEL / 16);
        int blocks = (tiles + 7) / 8;
        gemm_kernel<D_MODEL, D_MODEL, false><<<blocks, 256, 0, stream>>>(x, Wq, bq, nullptr, q, N);
        gemm_kernel<D_MODEL, D_MODEL, false><<<blocks, 256, 0, stream>>>(x, Wk, bk, nullptr, k, N);
        gemm_kernel<D_MODEL, D_MODEL, false><<<blocks, 256, 0, stream>>>(x, Wv, bv, nullptr, v, N);
    }
    // 4. ks_sum = column sums of k
    colsum_kernel<<<512, 128, 0, stream>>>(k, ks_sum, N);
    // 5. denom[n,h]
    denom_kernel<<<(N * NHEAD + 255) / 256, 256, 0, stream>>>(q, ks_sum, denom, N);
    // 6. edge phase: attn + atomic segment-sum into agg
    edge_kernel<<<(E + 7) / 8, 256, 0, stream>>>(eidx, spv, erel, rel_emb, sp_emb,
                                                 q, k, v, denom, agg, attn, E);
    // 7. out1 = agg @ Wd + bd + feature (WMMA + residual)
    {
        int tiles  = (N / 16) * (D_MODEL / 16);
        int blocks = (tiles + 7) / 8;
        gemm_kernel<D_MODEL, D_MODEL, true><<<blocks, 256, 0, stream>>>(agg, Wd, bd, feature, out1, N);
    }
    // 8. hln = LN(out1)  (reuses x buffer)
    ln_kernel<<<(N + 7) / 8, 256, 0, stream>>>(out1, g2, be2, x, N);
    // 9. fused FFN + residual -> out2 (WMMA, LDS-staged intermediate)
    ffn_kernel<<<N / 16, 256, 0, stream>>>(x, W1, b1, W2, b2, out1, out2);
}